// ParallelLinearRecurrence_28930899706261
// MI455X (gfx1250) — compile-verified
//
#include <hip/hip_runtime.h>
#include <cstddef>
#include <cstdint>

typedef __attribute__((ext_vector_type(2))) float v2f;
typedef __attribute__((ext_vector_type(8))) float v8f;

#define D_MODEL 256
#define SEQ_T   2048
#define BATCH   4
#define HEADS   4
#define DH      64
#define ROWS    (BATCH * SEQ_T)   // 8192
#define CHUNK   16
#define NCHUNK  (SEQ_T / CHUNK)   // 128

// f32 WMMA: D(16x16,f32) = A(16x4,f32) x B(4x16,f32) + C
// A-frag (v2f): lane L (l=L&15, half=L>>4), vgpr j -> A[M=l][K = k0 + j + 2*half]
// B-frag (v2f): vgpr j -> B[K = k0 + j + 2*half][N = l]
// C/D  (v8f):  vgpr r -> D[M = r + 8*half][N = l]
__device__ __forceinline__ v8f wmma4(v2f a, v2f b, v8f c) {
  return __builtin_amdgcn_wmma_f32_16x16x4_f32(
      /*neg_a=*/false, a, /*neg_b=*/false, b,
      /*c_mod=*/(short)0, c, /*reuse_a=*/false, /*reuse_b=*/false);
}

// Async global->LDS copy of 16 bytes (CDNA5 GLOBAL_LOAD_ASYNC_TO_LDS_B128,
// tracked by ASYNCcnt). ldsoff = wave-relative LDS byte offset (low 32 bits of
// a generic pointer to __shared__), gptr = 64-bit global address.
__device__ __forceinline__ void async_copy_b128(uint32_t ldsoff, const void* gptr) {
  asm volatile("global_load_async_to_lds_b128 %0, %1, off"
               :: "v"(ldsoff), "v"(gptr) : "memory");
}
__device__ __forceinline__ void async_wait0() {
  asm volatile("s_wait_asynccnt 0x0" ::: "memory");
}
__device__ __forceinline__ uint32_t lds_off(const void* p) {
  return (uint32_t)(uintptr_t)p;   // low 32 bits of flat shared addr = LDS offset
}

// Y[8192 x 256] = act(X @ W + bias); act: 0 = identity, 1 = elu(x)+1
__global__ __launch_bounds__(32)
void gemm_bias_act(const float* __restrict__ X, const float* __restrict__ W,
                   const float* __restrict__ bias, float* __restrict__ Y, int act) {
  const int m0   = blockIdx.x * 16;
  const int n0   = blockIdx.y * 16;
  const int lane = threadIdx.x & 31;
  const int half = lane >> 4;
  const int l    = lane & 15;

  v8f acc = {};
  const float* xrow = X + (size_t)(m0 + l) * D_MODEL;
  for (int k = 0; k < D_MODEL; k += 4) {
    const int kk = k + 2 * half;
    v2f a; a.x = xrow[kk];
           a.y = xrow[kk + 1];
    v2f b; b.x = W[(size_t)kk * D_MODEL + n0 + l];
           b.y = W[(size_t)(kk + 1) * D_MODEL + n0 + l];
    acc = wmma4(a, b, acc);
  }

  const float bn = bias[n0 + l];
  for (int r = 0; r < 8; ++r) {
    const int m = m0 + r + 8 * half;
    float v = acc[r] + bn;
    if (act) v = (v > 0.f) ? (v + 1.f) : __expf(v);   // elu(v)+1
    Y[(size_t)m * D_MODEL + n0 + l] = v;
  }
}

// Chunked causal linear attention. One workgroup (4 waves) per (b,h).
// Wave w owns output/state columns [16w, 16w+16) of dh=64.
__global__ __launch_bounds__(128)
void lin_attn_scan(const float* __restrict__ Q, const float* __restrict__ K,
                   const float* __restrict__ V, float* __restrict__ O) {
  __shared__ float Qs[CHUNK][DH];
  __shared__ float Ks[CHUNK][DH];
  __shared__ float Vs[CHUNK][DH];
  __shared__ float Slds[DH][DH];      // state S (K^T V cumulative) mirror
  __shared__ float Amask[CHUNK][CHUNK];
  __shared__ float zbuf[CHUNK][DH];   // inclusive cumsum of K within chunk (+carry)
  __shared__ float denom[CHUNK];

  const int b    = blockIdx.x >> 2;
  const int h    = blockIdx.x & 3;
  const int tid  = threadIdx.x;
  const int wave = tid >> 5;
  const int lane = tid & 31;
  const int half = lane >> 4;
  const int l    = lane & 15;
  const int n0   = wave * 16;

  const size_t base = (size_t)b * SEQ_T * D_MODEL + (size_t)h * DH;
  const float* Qb = Q + base;
  const float* Kb = K + base;
  const float* Vb = V + base;
  float*       Ob = O + base;

  // persistent state: wave w's 4 tiles of S[:, n0:n0+16] (d-blocks 0..3)
  v8f Sacc[4];
  for (int i = 0; i < 4; ++i) Sacc[i] = (v8f){};
  for (int i = tid; i < DH * DH; i += 128) Slds[i >> 6][i & 63] = 0.f;
  float zrun = 0.f;   // threads 0..63: running Z[d=tid]
  __syncthreads();

  for (int c = 0; c < NCHUNK; ++c) {
    const int t0 = c * CHUNK;

    // ---- stage chunk (16x64 each of Q,K,V) via async global->LDS b128 ----
    // 16*64 floats = 256 float4 per matrix; 128 threads -> 2 each.
    for (int i = tid; i < (CHUNK * DH) / 4; i += 128) {
      const int t = i >> 4, d4 = (i & 15) * 4;
      const size_t g = (size_t)(t0 + t) * D_MODEL + d4;
      async_copy_b128(lds_off(&Qs[t][d4]), Qb + g);
      async_copy_b128(lds_off(&Ks[t][d4]), Kb + g);
      async_copy_b128(lds_off(&Vs[t][d4]), Vb + g);
    }
    // prefetch next chunk's rows into cache while this chunk computes
    if (c + 1 < NCHUNK && tid < 48) {
      const int t = tid & 15, m = tid >> 4;
      const float* p = (m == 0 ? Qb : (m == 1 ? Kb : Vb)) +
                       (size_t)(t0 + CHUNK + t) * D_MODEL;
      __builtin_prefetch(p, 0, 3);
      __builtin_prefetch(p + 32, 0, 3);
    }
    async_wait0();
    __syncthreads();

    // ---- wave 0: A = Q_c K_c^T (16x16), causal mask (incl. diagonal) ----
    if (wave == 0) {
      v8f a = {};
      for (int k = 0; k < DH; k += 4) {
        const int kk = k + 2 * half;
        v2f af; af.x = Qs[l][kk];  af.y = Qs[l][kk + 1];
        v2f bf; bf.x = Ks[l][kk];  bf.y = Ks[l][kk + 1];   // B[K=d][N=t'] = K_c[t'][d]
        a = wmma4(af, bf, a);
      }
      for (int r = 0; r < 8; ++r) {
        const int row = r + 8 * half;                      // query t
        Amask[row][l] = (l <= row) ? a[r] : 0.f;           // keep t' <= t
      }
    }
    // ---- denominator path: inclusive cumsum of K over chunk ----
    if (tid < DH) {
      float z = zrun;
      for (int t = 0; t < CHUNK; ++t) { z += Ks[t][tid]; zbuf[t][tid] = z; }
      zrun = z;
    }
    __syncthreads();

    if (tid < CHUNK) {
      float s = 0.f;
      for (int d = 0; d < DH; ++d) s += Qs[tid][d] * zbuf[tid][d];
      denom[tid] = s + 1e-6f;
    }

    // ---- out tile = Amask @ V_c[:, n0:] + Q_c @ S_prev[:, n0:] ----
    v8f oacc = {};
    for (int k = 0; k < CHUNK; k += 4) {                   // intra-chunk
      const int kk = k + 2 * half;
      v2f af; af.x = Amask[l][kk];      af.y = Amask[l][kk + 1];
      v2f bf; bf.x = Vs[kk][n0 + l];    bf.y = Vs[kk + 1][n0 + l];
      oacc = wmma4(af, bf, oacc);
    }
    for (int k = 0; k < DH; k += 4) {                      // inter-chunk: Q @ S_prev
      const int kk = k + 2 * half;
      v2f af; af.x = Qs[l][kk];         af.y = Qs[l][kk + 1];
      v2f bf; bf.x = Slds[kk][n0 + l];  bf.y = Slds[kk + 1][n0 + l];
      oacc = wmma4(af, bf, oacc);
    }
    __syncthreads();   // all S_prev reads done; denom ready

    // ---- state update: S[:, n0:] += K_c^T @ V_c[:, n0:] ----
    for (int k = 0; k < CHUNK; k += 4) {
      const int kk = k + 2 * half;
      v2f bf; bf.x = Vs[kk][n0 + l];    bf.y = Vs[kk + 1][n0 + l];
      for (int i = 0; i < 4; ++i) {                        // d-blocks
        v2f af; af.x = Ks[kk][16 * i + l];  af.y = Ks[kk + 1][16 * i + l];
        Sacc[i] = wmma4(af, bf, Sacc[i]);
      }
    }
    for (int i = 0; i < 4; ++i)
      for (int r = 0; r < 8; ++r)
        Slds[16 * i + r + 8 * half][n0 + l] = Sacc[i][r];

    // ---- normalize and store output rows of this chunk ----
    for (int r = 0; r < 8; ++r) {
      const int t = r + 8 * half;
      Ob[(size_t)(t0 + t) * D_MODEL + n0 + l] = oacc[r] / denom[t];
    }
    __syncthreads();   // Slds/denom stable before next chunk overwrites staging
  }
}

extern "C" void kernel_launch(void* const* d_in, const int* in_sizes, int n_in,
                              void* d_out, int out_size, void* d_ws, size_t ws_size,
                              hipStream_t stream) {
  const float* x  = (const float*)d_in[0];
  const float* Wq = (const float*)d_in[1];
  const float* bq = (const float*)d_in[2];
  const float* Wk = (const float*)d_in[3];
  const float* bk = (const float*)d_in[4];
  const float* Wv = (const float*)d_in[5];
  const float* bv = (const float*)d_in[6];
  const float* Wo = (const float*)d_in[7];
  const float* bo = (const float*)d_in[8];
  float* out = (float*)d_out;

  const size_t n = (size_t)ROWS * D_MODEL;   // 2M elements per buffer
  float* Qbuf = (float*)d_ws;                // needs 4*n*4 = 32 MB of d_ws
  float* Kbuf = Qbuf + n;
  float* Vbuf = Kbuf + n;
  float* Obuf = Vbuf + n;

  dim3 g(ROWS / 16, D_MODEL / 16), blk(32);
  gemm_bias_act<<<g, blk, 0, stream>>>(x, Wq, bq, Qbuf, 1);  // elu+1
  gemm_bias_act<<<g, blk, 0, stream>>>(x, Wk, bk, Kbuf, 1);  // elu+1
  gemm_bias_act<<<g, blk, 0, stream>>>(x, Wv, bv, Vbuf, 0);
  lin_attn_scan<<<BATCH * HEADS, 128, 0, stream>>>(Qbuf, Kbuf, Vbuf, Obuf);
  gemm_bias_act<<<g, blk, 0, stream>>>(Obuf, Wo, bo, out, 0);
}